// LEIterator_16767552324128
// MI455X (gfx1250) — compile-verified
//
#include <hip/hip_runtime.h>
#include <cstdint>
#include <cmath>

#define S_TOTAL 1024
#define Q_NU    256
#define Q_1     128
#define NTRIP   44
#define KP      256
#define PMAX    49
#define SPB     8        // samples per block (double-buffered)
#define ROWLEN  65536    // output floats per sample

// ---------------------------------------------------------------------------
// Tables passed through kernarg (read-only). We read them straight from the
// kernarg segment (addrspace 4) to avoid the byval private-memory copy.
// sizeof(Tables) = 31504 bytes (8-aligned), placed at kernarg offset 0.
// ---------------------------------------------------------------------------
struct Tables {
  uint8_t idx_nu[NTRIP][KP];   // 0..255
  uint8_t idx_1[NTRIP][KP];    // 0..127
  float   cg[NTRIP][PMAX];
  int     outBase[NTRIP];      // offL[L] + t_order*KP
  int     segStride[NTRIP];    // nT[L]*KP
};

// ---------------------------------------------------------------------------
// Host: exact replication of numpy legacy RandomState(0) stream
// ---------------------------------------------------------------------------
namespace {

struct NpRng {
  uint32_t mt[624];
  int mti;
  bool has_gauss;
  double gauss;
  void seed(uint32_t s) {
    mt[0] = s;
    for (int i = 1; i < 624; ++i)
      mt[i] = 1812433253u * (mt[i - 1] ^ (mt[i - 1] >> 30)) + (uint32_t)i;
    mti = 624; has_gauss = false; gauss = 0.0;
  }
  uint32_t next() {
    if (mti >= 624) {
      for (int i = 0; i < 624; ++i) {
        uint32_t y = (mt[i] & 0x80000000u) | (mt[(i + 1) % 624] & 0x7fffffffu);
        mt[i] = mt[(i + 397) % 624] ^ (y >> 1) ^ ((y & 1u) ? 2567483615u : 0u);
      }
      mti = 0;
    }
    uint32_t y = mt[mti++];
    y ^= y >> 11;
    y ^= (y << 7)  & 2636928640u;
    y ^= (y << 15) & 4022730752u;
    y ^= y >> 18;
    return y;
  }
  double rk_double() {
    uint32_t a = next() >> 5, b = next() >> 6;
    return (a * 67108864.0 + b) / 9007199254740992.0;
  }
  double rk_gauss() {
    if (has_gauss) { has_gauss = false; return gauss; }
    double x1, x2, r2;
    do {
      x1 = 2.0 * rk_double() - 1.0;
      x2 = 2.0 * rk_double() - 1.0;
      r2 = x1 * x1 + x2 * x2;
    } while (r2 >= 1.0 || r2 == 0.0);
    double f = sqrt(-2.0 * log(r2) / r2);
    gauss = x1 * f; has_gauss = true;
    return x2 * f;
  }
};

void build_tables(Tables& tb) {
  int lamv[NTRIP], lv[NTRIP], Lv[NTRIP], ordv[NTRIP];
  int nT[7] = {0, 0, 0, 0, 0, 0, 0};
  int t = 0;
  for (int lam = 0; lam <= 3; ++lam)
    for (int l = 0; l <= 3; ++l)
      for (int L = (lam > l ? lam - l : l - lam); L <= lam + l; ++L) {
        lamv[t] = lam; lv[t] = l; Lv[t] = L; ordv[t] = nT[L]++; ++t;
      }
  int offL[7]; int acc = 0;
  for (int L = 0; L < 7; ++L) { offL[L] = acc; acc += (2 * L + 1) * nT[L] * KP; }

  NpRng rng; rng.seed(0);
  for (int tt = 0; tt < NTRIP; ++tt) {
    const int lam = lamv[tt], l = lv[tt], L = Lv[tt];
    for (int j = 0; j < KP; ++j) tb.idx_nu[tt][j] = (uint8_t)(rng.next() & (Q_NU - 1));
    for (int j = 0; j < KP; ++j) tb.idx_1[tt][j]  = (uint8_t)(rng.next() & (Q_1 - 1));
    int P = 0;
    for (int mu = 0; mu < 2 * lam + 1; ++mu)
      for (int m = 0; m < 2 * l + 1; ++m) {
        int Mv = (mu - lam) + (m - l);
        if (Mv >= -L && Mv <= L) ++P;
      }
    for (int q = 0; q < P; ++q)    tb.cg[tt][q] = (float)rng.rk_gauss();
    for (int q = P; q < PMAX; ++q) tb.cg[tt][q] = 0.0f;
    tb.outBase[tt]   = offL[L] + ordv[tt] * KP;
    tb.segStride[tt] = nT[L] * KP;
  }
}

} // namespace

// ---------------------------------------------------------------------------
// Device: CDNA5 async global->LDS staging (ASYNCcnt), double buffered
// ---------------------------------------------------------------------------
__device__ __forceinline__ void async_copy16(uint32_t lds_byte_addr, const float* g) {
  // LDS byte address taken from the vdst VGPR; 16B per active lane, tracked by ASYNCcnt
  asm volatile("global_load_async_to_lds_b128 %0, %1, off"
               :: "v"(lds_byte_addr), "v"(g) : "memory");
}
__device__ __forceinline__ void wait_async0() {
  asm volatile("s_wait_asynccnt 0" ::: "memory");
}

template <int AW, int BW>
__device__ __forceinline__ void stage(const float* __restrict__ gA,
                                      const float* __restrict__ gB,
                                      float* bufA, float* bufB, int which) {
  const int t = threadIdx.x;
  constexpr int CA = AW / 4;   // 16B chunks
  constexpr int CB = BW / 4;
  const uint32_t ldsA = (uint32_t)(uintptr_t)bufA + (uint32_t)which * (uint32_t)(AW * 4);
  const uint32_t ldsB = (uint32_t)(uintptr_t)bufB + (uint32_t)which * (uint32_t)(BW * 4);
#pragma unroll
  for (int c0 = 0; c0 < CA; c0 += 256) {
    const int idx = c0 + t;
    if (idx < CA) async_copy16(ldsA + (uint32_t)idx * 16u, gA + idx * 4);
  }
#pragma unroll
  for (int c0 = 0; c0 < CB; c0 += 256) {
    const int idx = c0 + t;
    if (idx < CB) async_copy16(ldsB + (uint32_t)idx * 16u, gB + idx * 4);
  }
}

template <int LAM, int LSM, int LL, int ID>
__device__ __forceinline__ void run_trip(const float* __restrict__ A,
                                         const float* __restrict__ B,
                                         float* __restrict__ out,
                                         const Tables& tb, float* smem) {
  constexpr int NA = 2 * LAM + 1;
  constexpr int NB = 2 * LSM + 1;
  constexpr int NO = 2 * LL + 1;
  constexpr int AW = NA * Q_NU;   // floats per sample (contiguous)
  constexpr int BW = NB * Q_1;

  float* bufA = smem;             // [2][AW]
  float* bufB = smem + 2 * AW;    // [2][BW]

  const int k     = threadIdx.x;
  const int my_nu = (int)tb.idx_nu[ID][k];
  const int my_1  = (int)tb.idx_1[ID][k];
  const int base  = tb.outBase[ID];
  const int seg   = tb.segStride[ID];
  const int s0    = blockIdx.y * SPB;

  stage<AW, BW>(A + (size_t)s0 * AW, B + (size_t)s0 * BW, bufA, bufB, 0);
  wait_async0();
  __syncthreads();

  for (int i = 0; i < SPB; ++i) {
    const int s = s0 + i;
    if (i + 1 < SPB)  // prefetch next sample into the other buffer
      stage<AW, BW>(A + (size_t)(s + 1) * AW, B + (size_t)(s + 1) * BW,
                    bufA, bufB, (i + 1) & 1);

    const float* a = bufA + (i & 1) * AW;
    const float* b = bufB + (i & 1) * BW;

    float Av[NA], Bv[NB], acc[NO];
#pragma unroll
    for (int mu = 0; mu < NA; ++mu) Av[mu] = a[mu * Q_NU + my_nu];  // LDS gather
#pragma unroll
    for (int m = 0; m < NB; ++m)    Bv[m]  = b[m * Q_1 + my_1];
#pragma unroll
    for (int M = 0; M < NO; ++M)    acc[M] = 0.0f;

    {
      int p = 0;  // fully unrolled -> all indices compile-time constants
#pragma unroll
      for (int mu = 0; mu < NA; ++mu)
#pragma unroll
        for (int m = 0; m < NB; ++m) {
          const int Mv = (mu - LAM) + (m - LSM);
          if (Mv >= -LL && Mv <= LL) {
            acc[Mv + LL] = fmaf(tb.cg[ID][p] * Av[mu], Bv[m], acc[Mv + LL]);
            ++p;
          }
        }
    }

    float* op = out + (size_t)s * ROWLEN + base + k;   // coalesced over k
#pragma unroll
    for (int M = 0; M < NO; ++M) op[(size_t)M * seg] = acc[M];

    wait_async0();     // prefetch of next buffer complete
    __syncthreads();   // all waves done reading current buffer
  }
}

// (id, lam, l, L) in reference generation order
#define TRIP_LIST(X) \
  X(0,0,0,0)  X(1,0,1,1)  X(2,0,2,2)  X(3,0,3,3) \
  X(4,1,0,1)  X(5,1,1,0)  X(6,1,1,1)  X(7,1,1,2) \
  X(8,1,2,1)  X(9,1,2,2)  X(10,1,2,3) \
  X(11,1,3,2) X(12,1,3,3) X(13,1,3,4) \
  X(14,2,0,2) X(15,2,1,1) X(16,2,1,2) X(17,2,1,3) \
  X(18,2,2,0) X(19,2,2,1) X(20,2,2,2) X(21,2,2,3) X(22,2,2,4) \
  X(23,2,3,1) X(24,2,3,2) X(25,2,3,3) X(26,2,3,4) X(27,2,3,5) \
  X(28,3,0,3) X(29,3,1,2) X(30,3,1,3) X(31,3,1,4) \
  X(32,3,2,1) X(33,3,2,2) X(34,3,2,3) X(35,3,2,4) X(36,3,2,5) \
  X(37,3,3,0) X(38,3,3,1) X(39,3,3,2) X(40,3,3,3) X(41,3,3,4) X(42,3,3,5) X(43,3,3,6)

// NOTE: `tb` is the FIRST kernel parameter -> lives at kernarg offset 0.
// We never reference the named parameter (which would force a 31.5KB
// per-thread private copy); instead we read it through the kernarg segment
// pointer (addrspace 4, scalar-cacheable). The dead byval copy is eliminated.
__global__ __launch_bounds__(256) void cg_kernel(
    Tables tb,
    const float* __restrict__ A0, const float* __restrict__ A1,
    const float* __restrict__ A2, const float* __restrict__ A3,
    const float* __restrict__ B0, const float* __restrict__ B1,
    const float* __restrict__ B2, const float* __restrict__ B3,
    float* __restrict__ out) {
  extern __shared__ float smem[];
  const char* karg = (const char*)__builtin_amdgcn_kernarg_segment_ptr();
  const Tables& tbr = *(const Tables*)karg;   // offset 0
  switch (blockIdx.x) {
#define CASE(ID, LAM, LSM, LL) \
    case ID: run_trip<LAM, LSM, LL, ID>(A##LAM, B##LSM, out, tbr, smem); break;
    TRIP_LIST(CASE)
#undef CASE
    default: break;
  }
}

// ---------------------------------------------------------------------------
// Launch
// ---------------------------------------------------------------------------
extern "C" void kernel_launch(void* const* d_in, const int* in_sizes, int n_in,
                              void* d_out, int out_size, void* d_ws, size_t ws_size,
                              hipStream_t stream) {
  (void)in_sizes; (void)n_in; (void)out_size; (void)d_ws; (void)ws_size;

  Tables tb;
  build_tables(tb);   // pure host math, deterministic, recomputed every call

  const float* A0 = (const float*)d_in[0];
  const float* A1 = (const float*)d_in[1];
  const float* A2 = (const float*)d_in[2];
  const float* A3 = (const float*)d_in[3];
  const float* B0 = (const float*)d_in[4];
  const float* B1 = (const float*)d_in[5];
  const float* B2 = (const float*)d_in[6];
  const float* B3 = (const float*)d_in[7];

  dim3 grid(NTRIP, S_TOTAL / SPB);
  dim3 block(256);
  const size_t shmem = (size_t)(2 * (7 * Q_NU) + 2 * (7 * Q_1)) * sizeof(float); // 21504 B

  cg_kernel<<<grid, block, shmem, stream>>>(tb, A0, A1, A2, A3, B0, B1, B2, B3,
                                            (float*)d_out);
}